// NonLocalNet_75969381532112
// MI455X (gfx1250) — compile-verified
//
#include <hip/hip_runtime.h>
#include <math.h>

#define B_SZ 4096
#define F_SZ 1024
#define P_SZ 6
#define D_SZ 128
#define O_SZ 512

typedef float v2f __attribute__((ext_vector_type(2)));
typedef float v8f __attribute__((ext_vector_type(8)));

__device__ __forceinline__ float lk(float x) { return x > 0.f ? x : 0.25f * x; }

// ---------------------------------------------------------------------------
// Pack embedding [B,F,P] -> [P,B,F] so per-part GEMM A-operands are contiguous.
// ---------------------------------------------------------------------------
__global__ void __launch_bounds__(256) pack_emb(const float* __restrict__ emb,
                                                float* __restrict__ embP) {
  const size_t idx = (size_t)blockIdx.x * 256 + threadIdx.x;  // over B*F
  const int b = (int)(idx >> 10);
  const int f = (int)(idx & 1023);
  const float* s = emb + ((size_t)b * F_SZ + f) * P_SZ;  // 6 contiguous floats
#pragma unroll
  for (int p = 0; p < P_SZ; ++p)
    embP[((size_t)p * B_SZ + b) * F_SZ + f] = s[p];
}

// ---------------------------------------------------------------------------
// Pack output [P,B,512] -> [B,512,P]
// ---------------------------------------------------------------------------
__global__ void __launch_bounds__(256) pack_out(const float* __restrict__ outP,
                                                float* __restrict__ out) {
  const size_t idx = (size_t)blockIdx.x * 256 + threadIdx.x;  // over B*O
  const int b = (int)(idx >> 9);
  const int o = (int)(idx & 511);
  float t[P_SZ];
#pragma unroll
  for (int p = 0; p < P_SZ; ++p)
    t[p] = outP[((size_t)p * B_SZ + b) * O_SZ + o];
  float* dst = out + ((size_t)b * O_SZ + o) * P_SZ;
#pragma unroll
  for (int p = 0; p < P_SZ; ++p) dst[p] = t[p];
}

// ---------------------------------------------------------------------------
// GEMM-NT via V_WMMA_F32_16X16X4_F32:  C[M,N] = A[M,K] * W[N,K]^T  (per part z)
//   block tile: 64 (M) x BN (N); wave tile: 64 x 64
//     -> 16 accumulators per wave; per K-chunk: 4 A-frag + 4 B-frag b64 LDS
//        reads feed 16 WMMAs (0.5 loads/WMMA).
//   LDS tiles padded to stride 36 words: conflict-free b64 fragment reads,
//   16B-aligned float4 staging.
// FUSE: A[b,f] = A[b,f] + leaky(U[b,f]*Sc[f] + Sh[f])  (residual + BN apply)
// LEAKY: epilogue LeakyReLU(0.25)
// ---------------------------------------------------------------------------
template <int BN, bool LEAKY, bool FUSE>
__global__ void __launch_bounds__(BN / 2)
gemm_nt(const float* __restrict__ A, const float* __restrict__ Wm,
        float* __restrict__ C, const float* __restrict__ U,
        const float* __restrict__ Sc, const float* __restrict__ Sh,
        int M, int N, int K, long sA, long sW, long sC) {
  constexpr int KB = 32;
  constexpr int KP = KB + 4;   // padded LDS row stride (words)
  constexpr int MR = 64;       // block tile rows
  constexpr int THREADS = BN / 2;
  __shared__ float As[MR * KP];
  __shared__ float Ws[BN * KP];

  const int p = blockIdx.z;
  A += (size_t)p * sA;
  Wm += (size_t)p * sW;
  C += (size_t)p * sC;
  const float* Up = FUSE ? (U + (size_t)p * sA) : nullptr;
  const float* Scp = FUSE ? (Sc + (size_t)p * K) : nullptr;
  const float* Shp = FUSE ? (Sh + (size_t)p * K) : nullptr;

  const int tid = threadIdx.x;
  const int wave = tid >> 5;
  const int lane = tid & 31;
  const int mn = lane & 15;  // A row / B col within 16x16 tile
  const int h = lane >> 4;   // lane-half: K offset 2*h for A/B fragments
  const int m0 = blockIdx.x * MR;
  const int n0 = blockIdx.y * BN;
  const int nw = wave * 64;

  v8f acc[4][4];  // [m-subtile][n-subtile]
#pragma unroll
  for (int ms = 0; ms < 4; ++ms)
#pragma unroll
    for (int t = 0; t < 4; ++t) acc[ms][t] = v8f{};

  for (int kb = 0; kb < K; kb += KB) {
    // --- stage A tile (64 x 32) ---
    for (int i = tid; i < MR * (KB / 4); i += THREADS) {
      const int r = i >> 3;
      const int c = (i & 7) * 4;
      const size_t g = (size_t)(m0 + r) * K + kb + c;
      float4 v = *(const float4*)(A + g);
      if constexpr (FUSE) {
        float4 uu = *(const float4*)(Up + g);
        float4 scv = *(const float4*)(Scp + kb + c);
        float4 shv = *(const float4*)(Shp + kb + c);
        v.x += lk(uu.x * scv.x + shv.x);
        v.y += lk(uu.y * scv.y + shv.y);
        v.z += lk(uu.z * scv.z + shv.z);
        v.w += lk(uu.w * scv.w + shv.w);
      }
      *(float4*)&As[r * KP + c] = v;
    }
    // --- stage W tile (BN x 32) ---
    for (int i = tid; i < BN * (KB / 4); i += THREADS) {
      const int r = i >> 3;
      const int c = (i & 7) * 4;
      float4 v = *(const float4*)(Wm + (size_t)(n0 + r) * K + kb + c);
      *(float4*)&Ws[r * KP + c] = v;
    }
    __syncthreads();

#pragma unroll
    for (int ch = 0; ch < KB / 4; ++ch) {
      const int ko = ch * 4 + 2 * h;  // K = 2*lanehalf + vgpr within 4-chunk
      v2f a[4], b[4];
#pragma unroll
      for (int ms = 0; ms < 4; ++ms)
        a[ms] = *(const v2f*)&As[(ms * 16 + mn) * KP + ko];
#pragma unroll
      for (int t = 0; t < 4; ++t)
        b[t] = *(const v2f*)&Ws[(nw + t * 16 + mn) * KP + ko];
#pragma unroll
      for (int ms = 0; ms < 4; ++ms)
#pragma unroll
        for (int t = 0; t < 4; ++t)
          acc[ms][t] = __builtin_amdgcn_wmma_f32_16x16x4_f32(
              false, a[ms], false, b[t], (short)0, acc[ms][t], false, false);
    }
    __syncthreads();
  }

  // --- store: C/D layout M = vgpr + 8*h, N = lane&15 ---
#pragma unroll
  for (int ms = 0; ms < 4; ++ms) {
#pragma unroll
    for (int t = 0; t < 4; ++t) {
#pragma unroll
      for (int r = 0; r < 8; ++r) {
        float v = acc[ms][t][r];
        if (LEAKY) v = lk(v);
        C[(size_t)(m0 + ms * 16 + r + 8 * h) * N + (n0 + nw + t * 16 + mn)] = v;
      }
    }
  }
}

// ---------------------------------------------------------------------------
// One wave (32 lanes) per batch row b: l2norm over D=128, 6x6 masked attention,
// column softmax (axis=1 == over p), emb_att = ps_n @ att.  Inputs/outputs in
// part-major workspace layout [P][B][D].
// ---------------------------------------------------------------------------
__global__ void __launch_bounds__(256)
attn_softmax(const float* __restrict__ ps, const float* __restrict__ cs,
             float* __restrict__ ea) {
  const int wave = threadIdx.x >> 5;
  const int lane = threadIdx.x & 31;
  const int b = blockIdx.x * 8 + wave;
  const size_t base = (size_t)b * D_SZ + lane * 4;

  float4 pl[P_SZ], cl[P_SZ];
#pragma unroll
  for (int p = 0; p < P_SZ; ++p) {
    pl[p] = *(const float4*)(ps + (size_t)p * B_SZ * D_SZ + base);
    cl[p] = *(const float4*)(cs + (size_t)p * B_SZ * D_SZ + base);
  }
  // l2 normalize over d (wave32 allreduce)
#pragma unroll
  for (int p = 0; p < P_SZ; ++p) {
    float s = pl[p].x * pl[p].x + pl[p].y * pl[p].y + pl[p].z * pl[p].z +
              pl[p].w * pl[p].w;
#pragma unroll
    for (int o = 16; o > 0; o >>= 1) s += __shfl_xor(s, o, 32);
    float rn = 1.f / (sqrtf(s) + 1e-8f);
    pl[p].x *= rn; pl[p].y *= rn; pl[p].z *= rn; pl[p].w *= rn;

    float c = cl[p].x * cl[p].x + cl[p].y * cl[p].y + cl[p].z * cl[p].z +
              cl[p].w * cl[p].w;
#pragma unroll
    for (int o = 16; o > 0; o >>= 1) c += __shfl_xor(c, o, 32);
    float rc = 1.f / (sqrtf(c) + 1e-8f);
    cl[p].x *= rc; cl[p].y *= rc; cl[p].z *= rc; cl[p].w *= rc;
  }
  // att[p][q] = <ps_n[:,p], cs_n[:,q]>  (replicated in every lane)
  float att[P_SZ][P_SZ];
#pragma unroll
  for (int p = 0; p < P_SZ; ++p) {
#pragma unroll
    for (int q = 0; q < P_SZ; ++q) {
      float s = pl[p].x * cl[q].x + pl[p].y * cl[q].y + pl[p].z * cl[q].z +
                pl[p].w * cl[q].w;
#pragma unroll
      for (int o = 16; o > 0; o >>= 1) s += __shfl_xor(s, o, 32);
      att[p][q] = s;
    }
  }
#pragma unroll
  for (int p = 0; p < P_SZ; ++p) att[p][p] += -1e6f;

  // softmax over p (per column q), then emb_att[:,q] = sum_p ps_n[:,p]*att[p][q]
#pragma unroll
  for (int q = 0; q < P_SZ; ++q) {
    float mx = att[0][q];
#pragma unroll
    for (int p = 1; p < P_SZ; ++p) mx = fmaxf(mx, att[p][q]);
    float sum = 0.f;
#pragma unroll
    for (int p = 0; p < P_SZ; ++p) {
      float e = __expf(att[p][q] - mx);
      att[p][q] = e;
      sum += e;
    }
    const float inv = 1.f / sum;
    float4 acc = make_float4(0.f, 0.f, 0.f, 0.f);
#pragma unroll
    for (int p = 0; p < P_SZ; ++p) {
      const float a = att[p][q] * inv;
      acc.x += pl[p].x * a;
      acc.y += pl[p].y * a;
      acc.z += pl[p].z * a;
      acc.w += pl[p].w * a;
    }
    *(float4*)(ea + (size_t)q * B_SZ * D_SZ + base) = acc;
  }
}

// ---------------------------------------------------------------------------
// Train-mode BN stats over B per (p,f): scale = gamma*rsqrt(var+eps),
// shift = beta - mean*scale.  Deterministic block reduction (no atomics).
// Block: 256 threads = (64 f-cols) x (4 b-stripes); grid: (F/64, P).
// ---------------------------------------------------------------------------
__global__ void __launch_bounds__(256)
bn_stats(const float* __restrict__ u, const float* __restrict__ gamma,
         const float* __restrict__ beta, float* __restrict__ scale,
         float* __restrict__ shift) {
  const int p = blockIdx.y;
  const int f0 = blockIdx.x * 64;
  const int tx = threadIdx.x & 63;
  const int ty = threadIdx.x >> 6;
  const float* up = u + (size_t)p * B_SZ * F_SZ;

  float s1 = 0.f, s2 = 0.f;
  for (int b = ty; b < B_SZ; b += 4) {
    const float v = up[(size_t)b * F_SZ + f0 + tx];
    s1 += v;
    s2 += v * v;
  }
  __shared__ float sh1[4][64], sh2[4][64];
  sh1[ty][tx] = s1;
  sh2[ty][tx] = s2;
  __syncthreads();
  if (ty == 0) {
    s1 = sh1[0][tx] + sh1[1][tx] + sh1[2][tx] + sh1[3][tx];
    s2 = sh2[0][tx] + sh2[1][tx] + sh2[2][tx] + sh2[3][tx];
    const float mean = s1 * (1.f / B_SZ);
    const float var = s2 * (1.f / B_SZ) - mean * mean;
    const float inv = rsqrtf(var + 1e-5f);
    const int idx = p * F_SZ + f0 + tx;
    const float sc = gamma[idx] * inv;
    scale[idx] = sc;
    shift[idx] = beta[idx] - mean * sc;
  }
}

// ---------------------------------------------------------------------------
extern "C" void kernel_launch(void* const* d_in, const int* in_sizes, int n_in,
                              void* d_out, int out_size, void* d_ws,
                              size_t ws_size, hipStream_t stream) {
  const float* emb = (const float*)d_in[0];
  const float* Wpart = (const float*)d_in[1];
  const float* Wcur = (const float*)d_in[2];
  const float* Wup = (const float*)d_in[3];
  const float* gamma = (const float*)d_in[4];
  const float* beta = (const float*)d_in[5];
  const float* Wlocal = (const float*)d_in[6];
  float* out = (float*)d_out;

  float* w = (float*)d_ws;
  float* embP = w;  w += (size_t)P_SZ * B_SZ * F_SZ;   // [P,B,F]
  float* ps = w;    w += (size_t)P_SZ * B_SZ * D_SZ;   // [P,B,D]
  float* cs = w;    w += (size_t)P_SZ * B_SZ * D_SZ;
  float* ea = w;    w += (size_t)P_SZ * B_SZ * D_SZ;
  float* u = w;     w += (size_t)P_SZ * B_SZ * F_SZ;   // [P,B,F]
  float* sc = w;    w += (size_t)P_SZ * F_SZ;
  float* sh = w;    w += (size_t)P_SZ * F_SZ;
  float* outP = w;  w += (size_t)P_SZ * B_SZ * O_SZ;   // [P,B,O]

  pack_emb<<<(B_SZ * F_SZ) / 256, 256, 0, stream>>>(emb, embP);

  // ps = leaky(embP x W_part^T), cs = leaky(embP x W_cur^T)   [N=128, K=1024]
  gemm_nt<128, true, false><<<dim3(B_SZ / 64, D_SZ / 128, P_SZ), 64, 0, stream>>>(
      embP, Wpart, ps, nullptr, nullptr, nullptr, B_SZ, D_SZ, F_SZ,
      (long)B_SZ * F_SZ, (long)D_SZ * F_SZ, (long)B_SZ * D_SZ);
  gemm_nt<128, true, false><<<dim3(B_SZ / 64, D_SZ / 128, P_SZ), 64, 0, stream>>>(
      embP, Wcur, cs, nullptr, nullptr, nullptr, B_SZ, D_SZ, F_SZ,
      (long)B_SZ * F_SZ, (long)D_SZ * F_SZ, (long)B_SZ * D_SZ);

  attn_softmax<<<B_SZ / 8, 256, 0, stream>>>(ps, cs, ea);

  // u = ea x W_up^T   [N=1024, K=128]
  gemm_nt<256, false, false><<<dim3(B_SZ / 64, F_SZ / 256, P_SZ), 128, 0, stream>>>(
      ea, Wup, u, nullptr, nullptr, nullptr, B_SZ, F_SZ, D_SZ,
      (long)B_SZ * D_SZ, (long)F_SZ * D_SZ, (long)B_SZ * F_SZ);

  bn_stats<<<dim3(F_SZ / 64, P_SZ), 256, 0, stream>>>(u, gamma, beta, sc, sh);

  // outP = (embP + leaky(u*sc+sh)) x W_local^T   [N=512, K=1024]
  gemm_nt<256, false, true><<<dim3(B_SZ / 64, O_SZ / 256, P_SZ), 128, 0, stream>>>(
      embP, Wlocal, outP, u, sc, sh, B_SZ, O_SZ, F_SZ,
      (long)B_SZ * F_SZ, (long)O_SZ * F_SZ, (long)B_SZ * O_SZ);

  pack_out<<<(B_SZ * O_SZ) / 256, 256, 0, stream>>>(outP, out);
}